// KNN_34119220199994
// MI455X (gfx1250) — compile-verified
//
#include <hip/hip_runtime.h>
#include <float.h>
#include <stdint.h>

typedef __attribute__((ext_vector_type(2))) float v2f;
typedef __attribute__((ext_vector_type(8))) float v8f;

#define KNN_N 8192
#define KNN_M 50000
#define KNN_D 128
#define KNN_K 10
#define KNN_C 10
#define TILES_M (KNN_M / 16)   /* 3125 */
#define WAVES 8
#define BROW 132               /* padded floats per B row (528B, 16B-aligned, conflict-free) */
#define BROWB (BROW * 4)

// Sorted-insert into ascending top-K list, guarded by current worst.
#define TOPK_INSERT(ksArr, kiArr, sVal, idVal)                                \
  do {                                                                        \
    if ((sVal) < ksArr[KNN_K - 1]) {                                          \
      ksArr[KNN_K - 1] = (sVal);                                              \
      kiArr[KNN_K - 1] = (idVal);                                             \
      _Pragma("unroll")                                                       \
      for (int _q = KNN_K - 1; _q > 0; --_q) {                                \
        if (ksArr[_q] < ksArr[_q - 1]) {                                      \
          float _ts = ksArr[_q]; ksArr[_q] = ksArr[_q-1]; ksArr[_q-1] = _ts;  \
          int   _ti = kiArr[_q]; kiArr[_q] = kiArr[_q-1]; kiArr[_q-1] = _ti;  \
        }                                                                     \
      }                                                                       \
    }                                                                         \
  } while (0)

// ---------------------------------------------------------------------------
// Kernel 1: y2[row] = sum_k train_pts[row][k]^2   (one wave = 8 rows, coalesced)
// ---------------------------------------------------------------------------
__global__ __launch_bounds__(256) void knn_y2_kernel(const float* __restrict__ tp,
                                                     float* __restrict__ y2) {
  const int lane = threadIdx.x & 31;
  const int wv   = threadIdx.x >> 5;
  const int wbase = (blockIdx.x * 8 + wv) * 8;
#pragma unroll
  for (int r = 0; r < 8; ++r) {
    const int row = wbase + r;
    if (row < KNN_M) {
      const float4 v = *(const float4*)(tp + (size_t)row * KNN_D + lane * 4);
      float s = v.x * v.x + v.y * v.y + v.z * v.z + v.w * v.w;
      s += __shfl_xor(s, 16);
      s += __shfl_xor(s, 8);
      s += __shfl_xor(s, 4);
      s += __shfl_xor(s, 2);
      s += __shfl_xor(s, 1);
      if (lane == 0) y2[row] = s;
    }
  }
}

// Issue one 16x128 f32 training tile as 16 async memory->LDS b128 copies.
// Each instruction moves 32 lanes x 16B = one 512B row; LDS dest uses the
// padded 528B row stride (async ops take independent global/LDS addresses).
__device__ __forceinline__ void issue_b_async(uint64_t gbase, uint32_t lbase) {
  // Async LDS writes are unordered vs this wave's DS reads: make sure all
  // prior ds_loads from the target buffer have completed.
  asm volatile("s_wait_dscnt 0x0" ::: "memory");
#pragma unroll
  for (int i = 0; i < 16; ++i) {
    asm volatile("global_load_async_to_lds_b128 %0, %1, off"
                 :: "v"(lbase + (uint32_t)(i * BROWB)),
                    "v"(gbase + (uint64_t)(i * (KNN_D * 4)))
                 : "memory");
  }
}

// ---------------------------------------------------------------------------
// Kernel 2: fused WMMA distance + online top-10 + vote.
// Block = 256 threads (8 waves) owning 16 query rows; waves split the 3125
// column tiles. score = ||y||^2 - 2 x.y (x^2 / sqrt are rank-invariant).
// B tiles are double-buffered in LDS via CDNA5 async global->LDS copies.
// ---------------------------------------------------------------------------
__global__ __launch_bounds__(256, 1) void knn_main_kernel(
    const float* __restrict__ x, const float* __restrict__ tp,
    const int* __restrict__ lab, const float* __restrict__ y2,
    float* __restrict__ out) {
  __shared__ float Bld[WAVES][2][16 * BROW];  // double-buffered train tiles
  __shared__ float Sld[WAVES][16 * 20];       // score tile transpose, stride 20
  __shared__ float Msc[WAVES * 16 * KNN_K];   // per-wave top-k merge buffers
  __shared__ int   Mid[WAVES * 16 * KNN_K];

  const int tid  = threadIdx.x;
  const int lane = tid & 31;
  const int wv   = tid >> 5;
  const int rowbase = blockIdx.x * 16;

  // WMMA f32 16x16x4 fragment addressing (wave32):
  //  A (16x4): lanes 0-15 -> M=lane, K={kb,kb+1}; lanes 16-31 -> M=lane-16, K={kb+2,kb+3}
  //  B (4x16): same pattern with N instead of M
  //  C (16x16): vgpr r -> M=r+8*(lane>=16), N=lane&15
  const int mn    = lane & 15;
  const int khalf = (lane >> 4) * 2;  // 0 or 2

  // ---- A fragments are tile-invariant: preload all 32 (64 VGPRs) ----
  v2f afrag[32];
  {
    const float* Ap = x + (size_t)(rowbase + mn) * KNN_D + khalf;
#pragma unroll
    for (int kk = 0; kk < 32; ++kk) afrag[kk] = *(const v2f*)(Ap + kk * 4);
  }

  float ks[KNN_K]; int ki[KNN_K];
#pragma unroll
  for (int q = 0; q < KNN_K; ++q) { ks[q] = FLT_MAX; ki[q] = 0; }

  float* const b0p = &Bld[wv][0][0];
  float* const b1p = &Bld[wv][1][0];
  const uint32_t l0 = (uint32_t)(uintptr_t)b0p + (uint32_t)(lane * 16);
  const uint32_t l1 = (uint32_t)(uintptr_t)b1p + (uint32_t)(lane * 16);
  const uint64_t gl = (uint64_t)(uintptr_t)tp + (uint64_t)(lane * 16);
  float* Sw = Sld[wv];

  // prologue: start the wave's first tile into buffer 0
  issue_b_async(gl + (uint64_t)(wv * 16) * (KNN_D * 4), l0);

  int pb = 0;
  for (int jt = wv; jt < TILES_M; jt += WAVES) {
    const int colbase = jt * 16;
    const int njt = jt + WAVES;

    if (njt < TILES_M) {
      // overlap: fetch next tile into the other buffer, then wait for current
      issue_b_async(gl + (uint64_t)(njt * 16) * (KNN_D * 4), pb ? l0 : l1);
      asm volatile("s_wait_asynccnt 0x10" ::: "memory");  // first 16 (in-order) done
    } else {
      asm volatile("s_wait_asynccnt 0x0" ::: "memory");
    }
    const float* Bw = pb ? b1p : b0p;

    const float y2v = y2[colbase + mn];

    // ---- 16x16x128 f32 matmul: 32 x v_wmma_f32_16x16x4_f32, 2 acc chains ----
    v8f acc0 = {};
    v8f acc1 = {};
    const float* Bf = Bw + mn * BROW + khalf;
#pragma unroll
    for (int kk = 0; kk < 32; kk += 2) {
      const v2f bA = *(const v2f*)(Bf + kk * 4);
      const v2f bB = *(const v2f*)(Bf + (kk + 1) * 4);
      acc0 = __builtin_amdgcn_wmma_f32_16x16x4_f32(false, afrag[kk], false, bA,
                                                   (short)0, acc0, false, false);
      acc1 = __builtin_amdgcn_wmma_f32_16x16x4_f32(false, afrag[kk + 1], false, bB,
                                                   (short)0, acc1, false, false);
    }

    // score = y2 - 2*dot ; transpose tile through padded LDS
#pragma unroll
    for (int r = 0; r < 8; ++r) {
      const int row = r + (khalf << 2);  // +8 for lanes 16-31
      Sw[row * 20 + mn] = fmaf(-2.0f, acc0[r] + acc1[r], y2v);
    }
    __builtin_amdgcn_wave_barrier();
    asm volatile("s_wait_dscnt 0x0" ::: "memory");  // wave-synchronous LDS handoff

    // lanes 0-15: each owns one query row, scan its 16 candidates
    if (lane < 16) {
#pragma unroll
      for (int c4 = 0; c4 < 4; ++c4) {
        const float4 sv = *(const float4*)(&Sw[lane * 20 + c4 * 4]);
        float sa[4] = {sv.x, sv.y, sv.z, sv.w};
#pragma unroll
        for (int j = 0; j < 4; ++j) {
          const float s = sa[j];
          const int id = colbase + c4 * 4 + j;
          TOPK_INSERT(ks, ki, s, id);
        }
      }
    }
    __builtin_amdgcn_wave_barrier();
    pb ^= 1;
  }

  // ---- publish per-wave lists, merge, vote ----
  if (lane < 16) {
#pragma unroll
    for (int q = 0; q < KNN_K; ++q) {
      Msc[(wv * 16 + lane) * KNN_K + q] = ks[q];
      Mid[(wv * 16 + lane) * KNN_K + q] = ki[q];
    }
  }
  __syncthreads();

  if (tid < 16) {
    float fs[KNN_K]; int fi[KNN_K];
#pragma unroll
    for (int q = 0; q < KNN_K; ++q) { fs[q] = FLT_MAX; fi[q] = 0; }
    for (int w = 0; w < WAVES; ++w) {
#pragma unroll
      for (int q = 0; q < KNN_K; ++q) {
        const float s = Msc[(w * 16 + tid) * KNN_K + q];
        const int  id = Mid[(w * 16 + tid) * KNN_K + q];
        TOPK_INSERT(fs, fi, s, id);
      }
    }
    int counts[KNN_C];
#pragma unroll
    for (int c = 0; c < KNN_C; ++c) counts[c] = 0;
#pragma unroll
    for (int q = 0; q < KNN_K; ++q) {
      const int lb = lab[fi[q]];
#pragma unroll
      for (int c = 0; c < KNN_C; ++c) counts[c] += (lb == c) ? 1 : 0;
    }
    // ascending scan with >= : max count, ties -> largest label (matches reference)
    int best = -1, win = 0;
#pragma unroll
    for (int c = 0; c < KNN_C; ++c) {
      if (counts[c] >= best) { best = counts[c]; win = c; }
    }
    out[rowbase + tid] = (float)win;
  }
}

extern "C" void kernel_launch(void* const* d_in, const int* in_sizes, int n_in,
                              void* d_out, int out_size, void* d_ws, size_t ws_size,
                              hipStream_t stream) {
  const float* x   = (const float*)d_in[0];
  const float* tp  = (const float*)d_in[1];
  const int*   lab = (const int*)d_in[2];
  // d_in[3] is k (==10); baked in as KNN_K.
  float* y2  = (float*)d_ws;   // 50000 floats = 200 KB of scratch
  float* out = (float*)d_out;  // integer labels written as float per harness contract

  const int y2blocks = (KNN_M + 63) / 64;  // 64 rows per block (8 waves x 8 rows)
  knn_y2_kernel<<<y2blocks, 256, 0, stream>>>(tp, y2);
  knn_main_kernel<<<KNN_N / 16, 256, 0, stream>>>(x, tp, lab, y2, out);

  (void)in_sizes; (void)n_in; (void)out_size; (void)ws_size;
}